// Voxelizer_23862838296732
// MI455X (gfx1250) — compile-verified
//
#include <hip/hip_runtime.h>

typedef __attribute__((ext_vector_type(2))) float v2f;
typedef __attribute__((ext_vector_type(8))) float v8f;

#define NG   256        // gaussians
#define KDIM 12         // 10 monomial features padded to 12 (3 x K=4 WMMA steps)
#define GRID_DIM 128
#define NVOX (GRID_DIM*GRID_DIM*GRID_DIM)
#define ROW_BLOCKS (NVOX/16)        // 131072 blocks of 16 voxels
#define MAIN_BLOCKS 2048            // 2048*8 waves -> exactly 8 iterations/wave

// ---------------------------------------------------------------------------
// Kernel 1: per-gaussian coefficient vector w_n[12] such that
//   mahal(v,n) = q_v . w_n,  q_v = [c0^2,c1^2,c2^2,c0c1,c0c2,c1c2,c0,c1,c2,1,0,0]
// with C = R diag(1/(s^2+eps)) R^T, b = C p, const = p^T C p.
// Layout: w[n*12 + k]  (n-major; stride 12 floats => float2-aligned, and 12-dword
// stride is coprime-ish with 64 banks => conflict-free LDS reads later).
// ---------------------------------------------------------------------------
__global__ __launch_bounds__(256) void vox_prep(const float* __restrict__ pos,
                                                const float* __restrict__ scl,
                                                const float* __restrict__ rot,
                                                float* __restrict__ w) {
  int n = blockIdx.x * blockDim.x + threadIdx.x;
  if (n >= NG) return;
  float qw = rot[n*4+0], qx = rot[n*4+1], qy = rot[n*4+2], qz = rot[n*4+3];
  float rn = rsqrtf(qw*qw + qx*qx + qy*qy + qz*qz);
  qw *= rn; qx *= rn; qy *= rn; qz *= rn;
  float R00 = 1.f - 2.f*(qy*qy + qz*qz);
  float R01 = 2.f*(qx*qy - qw*qz);
  float R02 = 2.f*(qx*qz + qw*qy);
  float R10 = 2.f*(qx*qy + qw*qz);
  float R11 = 1.f - 2.f*(qx*qx + qz*qz);
  float R12 = 2.f*(qy*qz - qw*qx);
  float R20 = 2.f*(qx*qz - qw*qy);
  float R21 = 2.f*(qy*qz + qw*qx);
  float R22 = 1.f - 2.f*(qx*qx + qy*qy);
  float s0 = scl[n*3+0], s1 = scl[n*3+1], s2 = scl[n*3+2];
  float i0 = 1.f/(s0*s0 + 1e-8f);
  float i1 = 1.f/(s1*s1 + 1e-8f);
  float i2 = 1.f/(s2*s2 + 1e-8f);
  float C00 = R00*R00*i0 + R01*R01*i1 + R02*R02*i2;
  float C01 = R00*R10*i0 + R01*R11*i1 + R02*R12*i2;
  float C02 = R00*R20*i0 + R01*R21*i1 + R02*R22*i2;
  float C11 = R10*R10*i0 + R11*R11*i1 + R12*R12*i2;
  float C12 = R10*R20*i0 + R11*R21*i1 + R12*R22*i2;
  float C22 = R20*R20*i0 + R21*R21*i1 + R22*R22*i2;
  float p0 = pos[n*3+0], p1 = pos[n*3+1], p2 = pos[n*3+2];
  float b0 = C00*p0 + C01*p1 + C02*p2;
  float b1 = C01*p0 + C11*p1 + C12*p2;
  float b2 = C02*p0 + C12*p1 + C22*p2;
  float cn = p0*b0 + p1*b1 + p2*b2;
  float* wn = w + n*KDIM;
  wn[0] = C00;      wn[1] = C11;      wn[2]  = C22;
  wn[3] = 2.f*C01;  wn[4] = 2.f*C02;  wn[5]  = 2.f*C12;
  wn[6] = -2.f*b0;  wn[7] = -2.f*b1;  wn[8]  = -2.f*b2;
  wn[9] = cn;       wn[10] = 0.f;     wn[11] = 0.f;
}

// ---------------------------------------------------------------------------
// Kernel 2: WMMA main loop.
// One wave owns 16 voxels (one M=16 row block). For each of the 16 column
// tiles (16 gaussians each): 3 x v_wmma_f32_16x16x4_f32 accumulate the
// Mahalanobis distances, then exp2(-0.5*log2e * m)*density is accumulated
// per lane. Butterfly shuffle reduces the 16 columns, lanes 0/16 store b128.
//
// Fragment layouts (ISA 7.12.2, 32-bit operands):
//   A 16x4 : lane L -> M = L&15, VGPR j holds K = 2*(L>>4) + j
//   B 4x16 : lane L -> N = L&15, VGPR j holds K = 2*(L>>4) + j
//   C 16x16: lane L, VGPR i -> M = i + 8*(L>>4), N = L&15
// ---------------------------------------------------------------------------
__global__ __launch_bounds__(256) void vox_main(const float* __restrict__ w,
                                                const float* __restrict__ density,
                                                float* __restrict__ out) {
  __shared__ float lw[NG * KDIM];   // 12 KB coefficient table
  __shared__ float ld[NG];          // densities

  const int tid = threadIdx.x;
  for (int i = tid; i < NG * KDIM; i += 256) lw[i] = w[i];
  if (tid < NG) ld[tid] = density[tid];
  __syncthreads();

  const int lane = tid & 31;
  const int wave = tid >> 5;
  const int m    = lane & 15;       // row within 16-voxel block / column within tile
  const bool h   = lane >= 16;      // K-half select for A/B fragments

  const float EXP_SCALE = -0.72134752044448170368f;  // -0.5 * log2(e)
  const float SC = 2.0f / 127.0f;                    // linspace(-1,1,128) step

  int rb = blockIdx.x * 8 + wave;
  const int stride = gridDim.x * 8;

  for (; rb < ROW_BLOCKS; rb += stride) {           // wave-uniform loop: EXEC all-ones
    const int v  = rb * 16 + m;
    const int iz = v >> 14;                          // H*W = 16384
    const int iy = (v >> 7) & 127;
    const int ix = v & 127;
    const float c0 = (float)iz * SC - 1.0f;          // grid order is (z,y,x)
    const float c1 = (float)iy * SC - 1.0f;
    const float c2 = (float)ix * SC - 1.0f;

    // A fragments: this lane's two K-slice monomials per K=4 chunk.
    v2f a0, a1, a2;
    a0.x = h ? c2*c2 : c0*c0;   a0.y = h ? c0*c1 : c1*c1;   // K {0,1}|{2,3}
    a1.x = h ? c0    : c0*c2;   a1.y = h ? c1    : c1*c2;   // K {4,5}|{6,7}
    a2.x = h ? 0.f   : c2;      a2.y = h ? 0.f   : 1.f;     // K {8,9}|{10,11}

    float acc[8];
    #pragma unroll
    for (int i = 0; i < 8; ++i) acc[i] = 0.f;

    #pragma unroll
    for (int t = 0; t < 16; ++t) {                  // 16 column tiles of 16 gaussians
      const int n = t * 16 + m;
      const v2f* wp = (const v2f*)&lw[n * KDIM];    // 8-byte aligned (even index)
      const int hs = h ? 1 : 0;
      v2f b0 = wp[0 + hs];                          // K chunk 0
      v2f b1 = wp[2 + hs];                          // K chunk 1
      v2f b2 = wp[4 + hs];                          // K chunk 2

      v8f c = {0.f,0.f,0.f,0.f,0.f,0.f,0.f,0.f};
      c = __builtin_amdgcn_wmma_f32_16x16x4_f32(false, a0, false, b0, (short)0, c, false, false);
      c = __builtin_amdgcn_wmma_f32_16x16x4_f32(false, a1, false, b1, (short)0, c, false, false);
      c = __builtin_amdgcn_wmma_f32_16x16x4_f32(false, a2, false, b2, (short)0, c, false, false);

      const float dn = ld[n];                       // density of this lane's column
      #pragma unroll
      for (int i = 0; i < 8; ++i)
        acc[i] += __builtin_amdgcn_exp2f(c[i] * EXP_SCALE) * dn;
    }

    // Reduce the 16 columns. Masks 1..8 keep lanes 0-15 (rows 0-7) and
    // lanes 16-31 (rows 8-15) in separate reduction groups, matching C layout.
    #pragma unroll
    for (int mask = 1; mask <= 8; mask <<= 1) {
      #pragma unroll
      for (int i = 0; i < 8; ++i)
        acc[i] += __shfl_xor(acc[i], mask, 32);
    }

    if (lane == 0 || lane == 16) {
      float4 lo = make_float4(acc[0], acc[1], acc[2], acc[3]);
      float4 hi = make_float4(acc[4], acc[5], acc[6], acc[7]);
      float* dst = out + rb * 16 + (h ? 8 : 0);     // 64B-aligned row block
      ((float4*)dst)[0] = lo;
      ((float4*)dst)[1] = hi;
    }
  }
}

// ---------------------------------------------------------------------------
// Host-side launcher. Inputs: positions(768), scales(768), rotations(1024),
// density(256), D, H, W (ints, fixed at 128 per reference). Output: 128^3 f32.
// ---------------------------------------------------------------------------
extern "C" void kernel_launch(void* const* d_in, const int* in_sizes, int n_in,
                              void* d_out, int out_size, void* d_ws, size_t ws_size,
                              hipStream_t stream) {
  const float* positions = (const float*)d_in[0];
  const float* scales    = (const float*)d_in[1];
  const float* rotations = (const float*)d_in[2];
  const float* density   = (const float*)d_in[3];
  float* out = (float*)d_out;
  float* wtab = (float*)d_ws;            // NG*KDIM floats = 12 KB scratch

  vox_prep<<<1, 256, 0, stream>>>(positions, scales, rotations, wtab);
  vox_main<<<MAIN_BLOCKS, 256, 0, stream>>>(wtab, density, out);
}